// GCNModel_51393578664448
// MI455X (gfx1250) — compile-verified
//
#include <hip/hip_runtime.h>

// ---------------------------------------------------------------------------
// GCN (2x GCNConv + linear head) for MI455X / gfx1250, wave32.
// Dense GEMMs use V_WMMA_F32_16X16X4_F32 (full fp32, exact vs reference).
// Sparse aggregation uses coalesced float4 gathers + global_atomic_add_f32.
// ---------------------------------------------------------------------------

typedef float v2f __attribute__((ext_vector_type(2)));
typedef float v8f __attribute__((ext_vector_type(8)));

#define HID 128   // IN_F == HID_F == 128
#define NPRED 16

// ---------------- degree / normalization ----------------

__global__ void k_set_deg(float* deg, int n) {
  int i = blockIdx.x * blockDim.x + threadIdx.x;
  if (i < n) deg[i] = 1.0f;              // self-loop weight
}

__global__ void k_deg_scatter(const int* __restrict__ col,
                              const float* __restrict__ w,
                              float* deg, int E) {
  int e = blockIdx.x * blockDim.x + threadIdx.x;
  if (e < E) atomicAdd(&deg[col[e]], w[e]);
}

__global__ void k_dinv(float* deg, int n) {
  int i = blockIdx.x * blockDim.x + threadIdx.x;
  if (i < n) {
    float d = deg[i];
    deg[i] = (d > 0.0f) ? rsqrtf(d) : 0.0f;   // in place: deg -> dinv
  }
}

__global__ void k_norm(const int* __restrict__ row, const int* __restrict__ col,
                       const float* __restrict__ w, const float* __restrict__ dinv,
                       float* __restrict__ norm, int E) {
  int e = blockIdx.x * blockDim.x + threadIdx.x;
  if (e < E) norm[e] = dinv[row[e]] * w[e] * dinv[col[e]];
}

// ---------------- dense GEMM: C[n,128] = A[n,128] @ W[128,128] -------------
// One wave computes a 16-row x 128-col stripe: 8 accumulator tiles of 16x16.
// K-loop steps by 4 using V_WMMA_F32_16X16X4_F32.
//
// Fragment layouts (ISA 7.12.2, wave32):
//   A 16x4:  lanes 0-15  -> row M=l, VGPR0=K, VGPR1=K+1
//            lanes 16-31 -> row M=l, VGPR0=K+2, VGPR1=K+3
//   B 4x16:  lanes 0-15  -> col N=l, VGPR0=row K, VGPR1=row K+1
//            lanes 16-31 -> col N=l, VGPR0=row K+2, VGPR1=row K+3
//   C 16x16: VGPR r: lanes 0-15 -> (M=r,   N=l)
//                    lanes 16-31-> (M=r+8, N=l)
__global__ void k_gemm_wmma(const float* __restrict__ A,
                            const float* __restrict__ W,
                            float* __restrict__ C, int n) {
  int wid  = (blockIdx.x * blockDim.x + threadIdx.x) >> 5;  // global wave id
  int lane = threadIdx.x & 31;
  int row0 = wid * 16;
  if (row0 >= n) return;               // wave-uniform guard (EXEC all-1 inside)

  int half = lane >> 4;                // 0: lanes 0-15, 1: lanes 16-31
  int l    = lane & 15;

  v8f acc[8];
#pragma unroll
  for (int t = 0; t < 8; ++t) acc[t] = (v8f){0,0,0,0,0,0,0,0};

  const float* arow = A + (size_t)(row0 + l) * HID;

  for (int k = 0; k < HID; k += 4) {
    // A fragment: contiguous pair of K values per lane -> one b64 load
    v2f afrag = *(const v2f*)(arow + k + half * 2);
#pragma unroll
    for (int t = 0; t < 8; ++t) {
      int ncol = t * 16 + l;
      v2f bfrag;
      bfrag.x = W[(size_t)(k + half * 2)     * HID + ncol];
      bfrag.y = W[(size_t)(k + half * 2 + 1) * HID + ncol];
      acc[t] = __builtin_amdgcn_wmma_f32_16x16x4_f32(
          /*neg_a=*/false, afrag, /*neg_b=*/false, bfrag,
          /*c_mod=*/(short)0, acc[t], /*reuse_a=*/false, /*reuse_b=*/false);
    }
  }

#pragma unroll
  for (int t = 0; t < 8; ++t) {
#pragma unroll
    for (int r = 0; r < 8; ++r) {
      int m    = row0 + r + half * 8;
      int ncol = t * 16 + l;
      C[(size_t)m * HID + ncol] = acc[t][r];
    }
  }
}

// ---------------- edge aggregation: acc[col] += xw[row] * norm -------------
// One wave per edge; each lane moves float4 (128 floats / 32 lanes).
__global__ void k_scatter(const float* __restrict__ xw,
                          const int* __restrict__ row,
                          const int* __restrict__ col,
                          const float* __restrict__ norm,
                          float* __restrict__ acc, int E) {
  int e = ((blockIdx.x * blockDim.x + threadIdx.x) >> 5);
  if (e >= E) return;                  // wave-uniform
  int lane = threadIdx.x & 31;
  int r = row[e], c = col[e];
  float s = norm[e];
  const float4 v = *(const float4*)(xw + (size_t)r * HID + lane * 4);
  float* dst = acc + (size_t)c * HID + lane * 4;
  atomicAdd(dst + 0, v.x * s);
  atomicAdd(dst + 1, v.y * s);
  atomicAdd(dst + 2, v.z * s);
  atomicAdd(dst + 3, v.w * s);
}

// ---------------- epilogue: h = relu(acc + xw*dinv^2 + b), in place --------
__global__ void k_epilogue(float* __restrict__ acc,
                           const float* __restrict__ xw,
                           const float* __restrict__ dinv,
                           const float* __restrict__ b, int n) {
  int i = blockIdx.x * blockDim.x + threadIdx.x;     // over n * 32 float4s
  if (i >= n * (HID / 4)) return;
  int node = i >> 5;
  int f4   = i & 31;
  float s  = dinv[node];
  s = s * s;                                         // self-loop norm
  float4 a  = ((float4*)acc)[i];
  float4 xv = ((const float4*)xw)[i];
  float4 bv = ((const float4*)b)[f4];
  float4 o;
  o.x = a.x + xv.x * s + bv.x;  o.x = o.x > 0.0f ? o.x : 0.0f;
  o.y = a.y + xv.y * s + bv.y;  o.y = o.y > 0.0f ? o.y : 0.0f;
  o.z = a.z + xv.z * s + bv.z;  o.z = o.z > 0.0f ? o.z : 0.0f;
  o.w = a.w + xv.w * s + bv.w;  o.w = o.w > 0.0f ? o.w : 0.0f;
  ((float4*)acc)[i] = o;
}

// ---------------- output head: out[n,16] = h @ Wout^T + bout ---------------
__global__ void k_out_proj(const float* __restrict__ h,
                           const float* __restrict__ Wout,
                           const float* __restrict__ bout,
                           float* __restrict__ out, int n) {
  int idx = blockIdx.x * blockDim.x + threadIdx.x;   // n * 16
  if (idx >= n * NPRED) return;
  int node = idx >> 4;
  int p    = idx & 15;
  const float4* hr = (const float4*)(h + (size_t)node * HID);
  const float4* wr = (const float4*)(Wout + (size_t)p * HID);
  float s = 0.0f;
#pragma unroll 8
  for (int k = 0; k < HID / 4; ++k) {
    float4 a = hr[k], b = wr[k];
    s += a.x * b.x + a.y * b.y + a.z * b.z + a.w * b.w;
  }
  out[idx] = s + bout[p];
}

// ---------------------------------------------------------------------------

extern "C" void kernel_launch(void* const* d_in, const int* in_sizes, int n_in,
                              void* d_out, int out_size, void* d_ws, size_t ws_size,
                              hipStream_t stream) {
  const float* x    = (const float*)d_in[0];   // [n,128]
  const int*   ei   = (const int*)  d_in[1];   // [2,E]  (row-major: row ids, then col ids)
  const float* ew   = (const float*)d_in[2];   // [E]
  const float* W1   = (const float*)d_in[3];   // [128,128]
  const float* b1   = (const float*)d_in[4];   // [128]
  const float* W2   = (const float*)d_in[5];   // [128,128]
  const float* b2   = (const float*)d_in[6];   // [128]
  const float* Wout = (const float*)d_in[7];   // [16,128]
  const float* bout = (const float*)d_in[8];   // [16]
  float* out = (float*)d_out;

  const int n = in_sizes[0] / HID;             // 100000
  const int E = in_sizes[2];                   // 1600000
  const int* erow = ei;                        // edge_index[0]
  const int* ecol = ei + E;                    // edge_index[1]

  // Workspace layout (floats): dinv[n] | norm[E] | xw[n*128] | acc[n*128]
  float* dinv = (float*)d_ws;
  float* norm = dinv + n;
  float* xw   = norm + E;
  float* acc  = xw + (size_t)n * HID;

  const int B = 256;
  int gN  = (n + B - 1) / B;
  int gE  = (E + B - 1) / B;
  int nTiles = (n + 15) / 16;
  int gGemm  = (nTiles + 7) / 8;               // 8 waves per 256-thread block
  int gScat  = (E + 7) / 8;                    // 8 edge-waves per block
  int gEpi   = (n * (HID / 4) + B - 1) / B;
  int gOut   = (n * NPRED + B - 1) / B;

  // --- shared normalization (identical for both layers) ---
  k_set_deg    <<<gN, B, 0, stream>>>(dinv, n);
  k_deg_scatter<<<gE, B, 0, stream>>>(ecol, ew, dinv, E);
  k_dinv       <<<gN, B, 0, stream>>>(dinv, n);
  k_norm       <<<gE, B, 0, stream>>>(erow, ecol, ew, dinv, norm, E);

  // --- layer 1: h1 = relu(aggregate(x @ W1) + b1) -> acc ---
  k_gemm_wmma<<<gGemm, B, 0, stream>>>(x, W1, xw, n);
  hipMemsetAsync(acc, 0, (size_t)n * HID * sizeof(float), stream);
  k_scatter  <<<gScat, B, 0, stream>>>(xw, erow, ecol, norm, acc, E);
  k_epilogue <<<gEpi, B, 0, stream>>>(acc, xw, dinv, b1, n);

  // --- layer 2: h2 = relu(aggregate(h1 @ W2) + b2) -> acc (in place) ---
  k_gemm_wmma<<<gGemm, B, 0, stream>>>(acc, W2, xw, n);
  hipMemsetAsync(acc, 0, (size_t)n * HID * sizeof(float), stream);
  k_scatter  <<<gScat, B, 0, stream>>>(xw, erow, ecol, norm, acc, E);
  k_epilogue <<<gEpi, B, 0, stream>>>(acc, xw, dinv, b2, n);

  // --- output head ---
  k_out_proj<<<gOut, B, 0, stream>>>(acc, Wout, bout, out, n);
}